// GNN_16707422781832
// MI455X (gfx1250) — compile-verified
//
#include <hip/hip_runtime.h>
#include <hip/hip_bf16.h>

typedef __attribute__((ext_vector_type(2))) float v2f;
typedef __attribute__((ext_vector_type(8))) float v8f;

#define DIM 64

// ---------------------------------------------------------------------------
// Zero the aggregation buffer (d_ws is poisoned; must init every call).
// ---------------------------------------------------------------------------
__global__ void gnn_zero_kernel(float4* __restrict__ p, int n4) {
    int i = blockIdx.x * blockDim.x + threadIdx.x;
    if (i < n4) p[i] = make_float4(0.f, 0.f, 0.f, 0.f);
}

// ---------------------------------------------------------------------------
// Edge scatter: agg[dst[e], :] += x[src[e], :].
// 16 threads per edge, each handles 4 consecutive dims via a float4 gather +
// 4 f32 atomics (-> global_atomic_add_f32, L2 atomic units; feat & agg both
// fit in the 192MB L2, so this is L2-bound, not HBM-bound).
// ---------------------------------------------------------------------------
__global__ __launch_bounds__(256) void gnn_scatter_kernel(
    const float* __restrict__ x, const int* __restrict__ src,
    const int* __restrict__ dst, float* __restrict__ agg, int n_edges) {
    int t = blockIdx.x * blockDim.x + threadIdx.x;
    int e = t >> 4;
    if (e >= n_edges) return;
    int q = (t & 15) << 2;           // dim base: 0,4,...,60
    int s = src[e];
    int d = dst[e];
    const float4 v = *reinterpret_cast<const float4*>(x + (size_t)s * DIM + q);
    float* p = agg + (size_t)d * DIM + q;
    atomicAdd(p + 0, v.x);
    atomicAdd(p + 1, v.y);
    atomicAdd(p + 2, v.z);
    atomicAdd(p + 3, v.w);
}

// ---------------------------------------------------------------------------
// out[n, j] = act( sum_k A[n,k] * W[j,k] + bias[j] )   (A: [n_rows,64], W: [64,64])
// Full-precision CDNA5 WMMA: V_WMMA_F32_16X16X4_F32, one wave per 16-row tile.
//   A 16x4 f32 layout : lane(0-15)=M, half(lane>=16) selects K={0,1}|{2,3}
//   B 4x16 f32 layout : lane&15 = N, half selects K offset 0|2; B[k][n]=W[n][k]
//   C 16x16 f32 layout: VGPR r -> M = r + 8*half, N = lane&15
// ---------------------------------------------------------------------------
__global__ __launch_bounds__(256) void gnn_gemm_kernel(
    const float* __restrict__ A, const float* __restrict__ W,
    const float* __restrict__ bias, float* __restrict__ out,
    int n_rows, int apply_tanh) {
    int wave = (blockIdx.x * blockDim.x + threadIdx.x) >> 5;   // global wave id
    int lane = threadIdx.x & 31;
    int row0 = wave * 16;
    if (row0 >= n_rows) return;      // wave-uniform: EXEC stays all-ones for WMMA

    int half = lane >> 4;            // 0 or 1
    int l16  = lane & 15;

    // Per-lane A source row, offset by this half's K pair.
    const float* arow = A + (size_t)(row0 + l16) * DIM + 2 * half;

    for (int nt = 0; nt < 4; ++nt) {
        v8f c = {};
        // B[k][n] = W[n][k]; lane's N is l16, K pair offset by half.
        const float* wrow = W + (size_t)(nt * 16 + l16) * DIM + 2 * half;
        #pragma unroll
        for (int k = 0; k < DIM; k += 4) {
            v2f a, b;
            a.x = arow[k];  a.y = arow[k + 1];
            b.x = wrow[k];  b.y = wrow[k + 1];
            // 8 args: (neg_a, A, neg_b, B, c_mod, C, reuse_a, reuse_b)
            c = __builtin_amdgcn_wmma_f32_16x16x4_f32(
                false, a, false, b, (short)0, c, false, false);
        }
        float bv = bias[nt * 16 + l16];
        #pragma unroll
        for (int r = 0; r < 8; ++r) {
            float v = c[r] + bv;
            if (apply_tanh) v = tanhf(v);
            out[(size_t)(row0 + r + 8 * half) * DIM + nt * 16 + l16] = v;
        }
    }
}

// ---------------------------------------------------------------------------
// inputs (setup_inputs order): feat, src, dst, W1, b1, W2, b2
// d_ws layout: agg [n_nodes*64 f32] | h [n_nodes*64 f32]   (25.6 MB total)
// ---------------------------------------------------------------------------
extern "C" void kernel_launch(void* const* d_in, const int* in_sizes, int n_in,
                              void* d_out, int out_size, void* d_ws, size_t ws_size,
                              hipStream_t stream) {
    const float* feat = (const float*)d_in[0];
    const int*   src  = (const int*)d_in[1];
    const int*   dst  = (const int*)d_in[2];
    const float* W1   = (const float*)d_in[3];
    const float* b1   = (const float*)d_in[4];
    const float* W2   = (const float*)d_in[5];
    const float* b2   = (const float*)d_in[6];
    float* out = (float*)d_out;

    const int n_nodes = in_sizes[0] / DIM;
    const int n_edges = in_sizes[1];

    float* agg = (float*)d_ws;
    float* h   = agg + (size_t)n_nodes * DIM;

    const int n4    = n_nodes * (DIM / 4);            // float4 count of agg
    const int zgrid = (n4 + 255) / 256;
    const int sgrid = (n_edges * 16 + 255) / 256;     // 16 threads / edge
    const int waves = (n_nodes + 15) / 16;            // one wave per 16 rows
    const int ggrid = (waves + 7) / 8;                // 8 waves / 256-thread block

    // Layer 1
    gnn_zero_kernel<<<zgrid, 256, 0, stream>>>((float4*)agg, n4);
    gnn_scatter_kernel<<<sgrid, 256, 0, stream>>>(feat, src, dst, agg, n_edges);
    gnn_gemm_kernel<<<ggrid, 256, 0, stream>>>(agg, W1, b1, h, n_nodes, 1);

    // Layer 2
    gnn_zero_kernel<<<zgrid, 256, 0, stream>>>((float4*)agg, n4);
    gnn_scatter_kernel<<<sgrid, 256, 0, stream>>>(h, src, dst, agg, n_edges);
    gnn_gemm_kernel<<<ggrid, 256, 0, stream>>>(agg, W2, b2, out, n_nodes, 0);
}